// ViterbiLoss_10866267259542
// MI455X (gfx1250) — compile-verified
//
#include <hip/hip_runtime.h>
#include <stdint.h>

// CRF forward (ViterbiLoss): B=64, T=256, K=48. One workgroup per batch.
// Streams 9KB transition tiles with the Tensor Data Mover (double-buffered,
// TENSORcnt), does the logsumexp recurrence as an exp-domain matvec on
// v_wmma_f32_16x16x4_f32 (3 waves x 16 columns, 12 K-chunks of 4).

#define NB   64
#define NT   256
#define NK   48
#define NKK  2304           // 48*48 elements per tile (9216 bytes)
#define START_TAG 46
#define END_TAG   47

typedef __attribute__((ext_vector_type(2))) float        v2f;
typedef __attribute__((ext_vector_type(8))) float        v8f;
typedef __attribute__((ext_vector_type(4))) unsigned int v4u;
typedef __attribute__((ext_vector_type(8))) int          v8i;
typedef __attribute__((ext_vector_type(4))) int          v4i;

// Tensor DMA: copy one contiguous 2304-f32 tile (scores[b,t,:,:]) -> LDS.
// D# per CDNA5 ISA ch.8: group0 = {count|flags, lds_addr, global_addr, type=2},
// group1 = {data_size=4B, tensor_dim0=2304, tensor_dim1=1, tile_dim0=2304,
//           tile_dim1=1, tensor_dim0_stride=2304}.
__device__ __forceinline__ void tdm_load_tile(uint32_t lds_off, const float* gptr) {
  uint64_t ga = (uint64_t)(uintptr_t)gptr;
  v4u g0;
  g0[0] = 1u;                                   // count=1, user descriptor
  g0[1] = lds_off;                              // LDS byte address
  g0[2] = (uint32_t)ga;                         // global_addr[31:0]
  g0[3] = (uint32_t)((ga >> 32) & 0x01FFFFFFu)  // global_addr[56:32]
        | (2u << 30);                           // type = 2 ("image")
  v8i g1;
  g1[0] = (int)(2u << 16);                      // wg_mask=0, data_size=2 (4B)
  g1[1] = (int)((NKK & 0xFFFF) << 16);          // tensor_dim0[15:0]
  g1[2] = (int)((NKK >> 16) | (1 << 16));       // tensor_dim0[31:16] | tensor_dim1=1
  g1[3] = (int)(NKK << 16);                     // tile_dim0 = 2304
  g1[4] = 1;                                    // tile_dim1=1, tile_dim2=0
  g1[5] = NKK;                                  // tensor_dim0_stride[31:0]
  g1[6] = 0;
  g1[7] = 0;
  v4i z4 = {0, 0, 0, 0};
  v8i z8 = {0, 0, 0, 0, 0, 0, 0, 0};
  // clang-23 toolchain form: 6 args (g0, g1, g2, g3, extra, cpol)
  __builtin_amdgcn_tensor_load_to_lds(g0, g1, z4, z4, z8, 0);
}

__global__ __launch_bounds__(96) void crf_fwd_kernel(
    const float* __restrict__ S,          // [B,T,K,K]
    const long long* __restrict__ tgts,   // [B,T]
    const long long* __restrict__ lens,   // [B]
    float* __restrict__ ws)               // [B] per-batch loss
{
  __shared__ float tile[2][NKK];          // double-buffered transition tiles
  __shared__ float smPrev[NK];            // forward state (log domain)
  __shared__ float smRed[96];             // gold-score reduction

  const int b    = blockIdx.x;
  const int tid  = threadIdx.x;
  const int lane = tid & 31;
  const int wv   = tid >> 5;              // wave 0..2, owns columns 16w..16w+15

  const float* Sb = S + (size_t)b * NT * NKK;
  int len = (int)lens[b];
  if (len < 1)  len = 1;
  if (len > NT) len = NT;

  // Kick off the DMA of tile t=1 ASAP so it overlaps the gold-score pass.
  if (wv == 0 && len > 1) {
    tdm_load_tile((uint32_t)(uintptr_t)&tile[0][0], Sb + (size_t)NKK);
  }

  // ---- gold score: masked gather-sum of scores at target indices ----
  float g = 0.0f;
  for (int t = tid; t < NT; t += 96) {
    if (t < len) {
      int tg = (int)tgts[(size_t)b * NT + t];
      g += Sb[(size_t)t * NKK + tg];
    }
  }
  smRed[tid] = g;

  // ---- init: prev[j] = scores[b,0,START_TAG,j] ----
  if (tid < NK) smPrev[tid] = Sb[START_TAG * NK + tid];
  __syncthreads();

  float gold = 0.0f;
  if (tid == 0) {
    for (int i = 0; i < 96; ++i) gold += smRed[i];
  }

  const int jcol = wv * 16 + (lane & 15);       // this lane's output column
  const int koff = (lane < 16) ? 0 : 2;         // WMMA K-row pair per lane half

  // ---- forward recurrence over t = 1..len-1 ----
  for (int t = 1; t < len; ++t) {
    const int cur = (t - 1) & 1;
    if (wv == 0) {
      if (t + 1 < len) {
        tdm_load_tile((uint32_t)(uintptr_t)&tile[cur ^ 1][0],
                      Sb + (size_t)(t + 1) * NKK);
        __builtin_amdgcn_s_wait_tensorcnt(1);   // tile[cur] done, prefetch in flight
      } else {
        __builtin_amdgcn_s_wait_tensorcnt(0);
      }
    }
    __syncthreads();                            // tile[cur] visible to all waves

    // rescale constant: c = max_i prev[i] (exact logsumexp shift)
    float c = smPrev[0];
    #pragma unroll 12
    for (int i = 1; i < NK; ++i) c = fmaxf(c, smPrev[i]);

    // exp-domain matvec: acc[j] = sum_i exp(s[i,j]) * exp(prev[i]-c)
    // A = ep chunk broadcast over all 16 rows -> every D row equals acc.
    const float* tl = &tile[cur][0];
    v8f acc = {0.f, 0.f, 0.f, 0.f, 0.f, 0.f, 0.f, 0.f};
    #pragma unroll
    for (int ic = 0; ic < 12; ++ic) {
      const int r0 = ic * 4 + koff;
      v2f A, Bf;
      A[0]  = __expf(smPrev[r0]     - c);
      A[1]  = __expf(smPrev[r0 + 1] - c);
      Bf[0] = __expf(tl[r0       * NK + jcol]);
      Bf[1] = __expf(tl[(r0 + 1) * NK + jcol]);
      acc = __builtin_amdgcn_wmma_f32_16x16x4_f32(
          false, A, false, Bf, (short)0, acc, false, false);
    }
    float nv = __logf(acc[0]) + c;

    __syncthreads();                            // all reads of smPrev/tile done
    if (lane < 16) smPrev[jcol] = nv;           // 3 waves x 16 lanes cover 48 tags
    __syncthreads();                            // safe to prefetch into tile[cur]
  }

  if (tid == 0) ws[b] = smPrev[END_TAG] - gold;
}

__global__ void crf_reduce_kernel(const float* __restrict__ ws,
                                  float* __restrict__ out) {
  __shared__ float s[NB];
  s[threadIdx.x] = ws[threadIdx.x];
  __syncthreads();
  if (threadIdx.x == 0) {
    float a = 0.0f;
    for (int i = 0; i < NB; ++i) a += s[i];
    out[0] = a * (1.0f / NB);
  }
}

extern "C" void kernel_launch(void* const* d_in, const int* in_sizes, int n_in,
                              void* d_out, int out_size, void* d_ws, size_t ws_size,
                              hipStream_t stream) {
  (void)in_sizes; (void)n_in; (void)out_size; (void)ws_size;
  const float*     S  = (const float*)d_in[0];
  const long long* tg = (const long long*)d_in[1];
  const long long* ln = (const long long*)d_in[2];
  float* ws = (float*)d_ws;             // 64 floats of scratch
  crf_fwd_kernel<<<NB, 96, 0, stream>>>(S, tg, ln, ws);
  crf_reduce_kernel<<<1, NB, 0, stream>>>(ws, (float*)d_out);
}